// RingRiskGNN_90417651515945
// MI455X (gfx1250) — compile-verified
//
#include <hip/hip_runtime.h>

// ---------------------------------------------------------------------------
// RingRiskGNN: 2-layer GAT (heads=4,hidden=32 -> heads=1,hidden=32) +
// global mean pool + linear + sigmoid, on gfx1250 (CDNA5, wave32, WMMA).
// ---------------------------------------------------------------------------

typedef float v2f __attribute__((ext_vector_type(2)));
typedef float v8f __attribute__((ext_vector_type(8)));

#define NEG_SLOPE 0.2f

__device__ __forceinline__ float lrelu(float v) { return v > 0.f ? v : NEG_SLOPE * v; }

// Correct mixed-sign float atomic max via int/uint monotonic mapping.
// Memory must be initialized to -inf (0xFF800000).
__device__ __forceinline__ void atomicMaxF(float* addr, float val) {
  if (val >= 0.f) atomicMax((int*)addr, __float_as_int(val));
  else            atomicMin((unsigned int*)addr, __float_as_uint(val));
}

// Edge i in [0, E+N): i < E -> real edge, else self-loop (src=dst=i-E).
__device__ __forceinline__ void edge_sd(const int* __restrict__ ei, int i, int E,
                                        int& s, int& d) {
  if (i < E) { s = ei[i]; d = ei[E + i]; }
  else       { s = i - E; d = s; }
}

// --------------------------- small prep kernels ----------------------------

// ws1[h] = <W1[h,:], a_src1[h,:]>, wd1[h] = <W1[h,:], a_dst1[h,:]>  (h<4)
__global__ void prep1_kernel(const float* __restrict__ W1,
                             const float* __restrict__ a_src1,
                             const float* __restrict__ a_dst1,
                             float* __restrict__ ws1, float* __restrict__ wd1) {
  int t = threadIdx.x;
  if (t >= 8) return;
  int h = t & 3;
  const float* a = (t < 4) ? a_src1 : a_dst1;
  float acc = 0.f;
  #pragma unroll
  for (int c = 0; c < 32; ++c) acc += W1[h * 32 + c] * a[h * 32 + c];
  if (t < 4) ws1[h] = acc; else wd1[h] = acc;
}

__global__ void init_neginf_kernel(float* __restrict__ p, long long n) {
  long long i = (long long)blockIdx.x * blockDim.x + threadIdx.x;
  if (i < n) p[i] = -__builtin_inff();
}

// --------------------------- layer-1 edge passes ---------------------------
// e[edge,h] = leakyrelu(x[src]*ws1[h] + x[dst]*wd1[h])

__global__ void edge1_max_kernel(const int* __restrict__ ei, const float* __restrict__ x,
                                 const float* __restrict__ ws1, const float* __restrict__ wd1,
                                 float* __restrict__ m1, int E, int ET) {
  int i = blockIdx.x * blockDim.x + threadIdx.x;
  if (i >= ET) return;
  if (i + 4096 < 2 * E) __builtin_prefetch(ei + i + 4096, 0, 1);  // global_prefetch_b8
  int s, d; edge_sd(ei, i, E, s, d);
  float xs = x[s], xd = x[d];
  #pragma unroll
  for (int h = 0; h < 4; ++h) {
    float e = lrelu(xs * ws1[h] + xd * wd1[h]);
    atomicMaxF(&m1[d * 4 + h], e);
  }
}

__global__ void edge1_sum_kernel(const int* __restrict__ ei, const float* __restrict__ x,
                                 const float* __restrict__ ws1, const float* __restrict__ wd1,
                                 const float* __restrict__ m1, float* __restrict__ s1,
                                 int E, int ET) {
  int i = blockIdx.x * blockDim.x + threadIdx.x;
  if (i >= ET) return;
  int s, d; edge_sd(ei, i, E, s, d);
  float xs = x[s], xd = x[d];
  #pragma unroll
  for (int h = 0; h < 4; ++h) {
    float e = lrelu(xs * ws1[h] + xd * wd1[h]);
    atomicAdd(&s1[d * 4 + h], __expf(e - m1[d * 4 + h]));
  }
}

// Factorized aggregation: t1[dst,h] += alpha * x[src]  (out1 = t1[n,h]*W1[h,c])
__global__ void edge1_agg_kernel(const int* __restrict__ ei, const float* __restrict__ x,
                                 const float* __restrict__ ws1, const float* __restrict__ wd1,
                                 const float* __restrict__ m1, const float* __restrict__ s1,
                                 float* __restrict__ t1, int E, int ET) {
  int i = blockIdx.x * blockDim.x + threadIdx.x;
  if (i >= ET) return;
  int s, d; edge_sd(ei, i, E, s, d);
  float xs = x[s], xd = x[d];
  #pragma unroll
  for (int h = 0; h < 4; ++h) {
    float e = lrelu(xs * ws1[h] + xd * wd1[h]);
    float alpha = __expf(e - m1[d * 4 + h]) / (s1[d * 4 + h] + 1e-16f);
    atomicAdd(&t1[d * 4 + h], alpha * xs);
  }
}

// x1[n, h*32+c] = relu(t1[n,h]*W1[h*32+c] + b1[h*32+c]); one float4 per thread.
__global__ void build_x1_kernel(const float* __restrict__ t1, const float* __restrict__ W1,
                                const float* __restrict__ b1, float* __restrict__ x1, int N) {
  int tid = blockIdx.x * blockDim.x + threadIdx.x;
  int n = tid >> 5, q = tid & 31;            // q: float4 index in row of 128
  if (n >= N) return;
  int h = q >> 3;                            // 8 quads per head
  float t = t1[n * 4 + h];
  float4 w = ((const float4*)W1)[q];
  float4 bb = ((const float4*)b1)[q];
  float4 o;
  o.x = fmaxf(t * w.x + bb.x, 0.f);
  o.y = fmaxf(t * w.y + bb.y, 0.f);
  o.z = fmaxf(t * w.z + bb.z, 0.f);
  o.w = fmaxf(t * w.w + bb.w, 0.f);
  ((float4*)x1)[(long long)n * 32 + q] = o;
}

// ------------------------ layer-2 GEMM via f32 WMMA ------------------------
// h2[N,32] = x1[N,128] @ W2[128,32]; one wave -> 16x32 tile,
// two f32 accumulators, K stepped by 4 with V_WMMA_F32_16X16X4_F32.
__global__ void __launch_bounds__(256)
gemm_h2_kernel(const float* __restrict__ x1, const float* __restrict__ W2,
               float* __restrict__ h2, int N) {
  __shared__ float lw[4096];                 // W2 staged in LDS (16 KB)
  for (int i = threadIdx.x; i < 4096; i += blockDim.x) lw[i] = W2[i];
  __syncthreads();

  int wave = blockIdx.x * (blockDim.x >> 5) + (threadIdx.x >> 5);
  int lane = threadIdx.x & 31;
  int m0 = wave * 16;
  if (m0 >= N) return;

  int mrow = lane & 15;                      // A: M index / B: N index
  int kgrp = (lane >> 4) * 2;                // upper half-wave holds K=2,3

  int row = m0 + mrow; if (row >= N) row = N - 1;
  const float* arow = x1 + (long long)row * 128;

  v8f c0 = {}; v8f c1 = {};
  #pragma unroll 8
  for (int k = 0; k < 128; k += 4) {
    v2f a = *(const v2f*)(arow + k + kgrp);  // A[m, k+kgrp .. +1]
    v2f b0, b1f;                             // B[k+kgrp .. +1, n] / n+16
    b0.x  = lw[(k + kgrp)     * 32 + mrow];
    b0.y  = lw[(k + kgrp + 1) * 32 + mrow];
    b1f.x = lw[(k + kgrp)     * 32 + 16 + mrow];
    b1f.y = lw[(k + kgrp + 1) * 32 + 16 + mrow];
    c0 = __builtin_amdgcn_wmma_f32_16x16x4_f32(false, a, false, b0,  (short)0, c0, false, false);
    c1 = __builtin_amdgcn_wmma_f32_16x16x4_f32(false, a, false, b1f, (short)0, c1, false, false);
  }

  // C/D layout: VGPR v, lanes 0-15 -> M=v; lanes 16-31 -> M=v+8; N = lane&15.
  int nc = lane & 15;
  int mh = (lane >> 4) * 8;
  #pragma unroll
  for (int v = 0; v < 8; ++v) {
    int m = m0 + mh + v;
    if (m < N) {
      h2[(long long)m * 32 + nc]      = c0[v];
      h2[(long long)m * 32 + 16 + nc] = c1[v];
    }
  }
}

// es2[n] = <h2[n,:], a_src2>, ed2[n] = <h2[n,:], a_dst2>
__global__ void attn2_kernel(const float* __restrict__ h2, const float* __restrict__ a_src2,
                             const float* __restrict__ a_dst2,
                             float* __restrict__ es2, float* __restrict__ ed2, int N) {
  int n = blockIdx.x * blockDim.x + threadIdx.x;
  if (n >= N) return;
  const float4* hp = (const float4*)(h2 + (long long)n * 32);
  float as = 0.f, ad = 0.f;
  #pragma unroll
  for (int q = 0; q < 8; ++q) {
    float4 h = hp[q];
    float4 s4 = ((const float4*)a_src2)[q];
    float4 d4 = ((const float4*)a_dst2)[q];
    as += h.x * s4.x + h.y * s4.y + h.z * s4.z + h.w * s4.w;
    ad += h.x * d4.x + h.y * d4.y + h.z * d4.z + h.w * d4.w;
  }
  es2[n] = as; ed2[n] = ad;
}

// --------------------------- layer-2 edge passes ---------------------------

__global__ void edge2_max_kernel(const int* __restrict__ ei, const float* __restrict__ es2,
                                 const float* __restrict__ ed2, float* __restrict__ m2,
                                 int E, int ET) {
  int i = blockIdx.x * blockDim.x + threadIdx.x;
  if (i >= ET) return;
  int s, d; edge_sd(ei, i, E, s, d);
  atomicMaxF(&m2[d], lrelu(es2[s] + ed2[d]));
}

__global__ void edge2_sum_kernel(const int* __restrict__ ei, const float* __restrict__ es2,
                                 const float* __restrict__ ed2, const float* __restrict__ m2,
                                 float* __restrict__ s2, int E, int ET) {
  int i = blockIdx.x * blockDim.x + threadIdx.x;
  if (i >= ET) return;
  int s, d; edge_sd(ei, i, E, s, d);
  float e = lrelu(es2[s] + ed2[d]);
  atomicAdd(&s2[d], __expf(e - m2[d]));
}

// out2[dst,c] += alpha * h2[src,c]; 8 threads per edge, float4 channels.
__global__ void edge2_agg_kernel(const int* __restrict__ ei, const float* __restrict__ es2,
                                 const float* __restrict__ ed2, const float* __restrict__ m2,
                                 const float* __restrict__ s2, const float* __restrict__ h2,
                                 float* __restrict__ out2, int E, int ET) {
  int tid = blockIdx.x * blockDim.x + threadIdx.x;
  int i = tid >> 3, q = tid & 7;
  if (i >= ET) return;
  int s, d; edge_sd(ei, i, E, s, d);
  float e = lrelu(es2[s] + ed2[d]);
  float alpha = __expf(e - m2[d]) / (s2[d] + 1e-16f);
  float4 hv = ((const float4*)h2)[(long long)s * 8 + q];
  float* o = out2 + (long long)d * 32 + q * 4;
  atomicAdd(o + 0, alpha * hv.x);
  atomicAdd(o + 1, alpha * hv.y);
  atomicAdd(o + 2, alpha * hv.z);
  atomicAdd(o + 3, alpha * hv.w);
}

// x2 = relu(out2 + b2); mean-pool accumulation per graph.
__global__ void pool_kernel(const float* __restrict__ out2, const float* __restrict__ b2,
                            const int* __restrict__ batch, float* __restrict__ gsum,
                            float* __restrict__ gcnt, int N) {
  int tid = blockIdx.x * blockDim.x + threadIdx.x;
  int n = tid >> 5, c = tid & 31;
  if (n >= N) return;
  int g = batch[n];
  float v = fmaxf(out2[(long long)n * 32 + c] + b2[c], 0.f);
  atomicAdd(&gsum[g * 32 + c], v);
  if (c == 0) atomicAdd(&gcnt[g], 1.0f);
}

__global__ void final_kernel(const float* __restrict__ gsum, const float* __restrict__ gcnt,
                             const float* __restrict__ lin_w, const float* __restrict__ lin_b,
                             float* __restrict__ out, int G) {
  int g = blockIdx.x * blockDim.x + threadIdx.x;
  if (g >= G) return;
  float inv = 1.0f / fmaxf(gcnt[g], 1.0f);
  float acc = lin_b[0];
  #pragma unroll
  for (int c = 0; c < 32; ++c) acc += gsum[g * 32 + c] * inv * lin_w[c];
  out[g] = 1.0f / (1.0f + __expf(-acc));
}

// ------------------------------- launch ------------------------------------

static inline int cdiv_i(long long a, long long b) { return (int)((a + b - 1) / b); }
static inline long long align4(long long v) { return (v + 3) & ~3LL; }

extern "C" void kernel_launch(void* const* d_in, const int* in_sizes, int n_in,
                              void* d_out, int out_size, void* d_ws, size_t ws_size,
                              hipStream_t stream) {
  const float* x      = (const float*)d_in[0];
  const int*   ei     = (const int*)d_in[1];     // [2,E]
  const int*   batch  = (const int*)d_in[2];     // [N]
  // d_in[3] = num_graphs scalar (G also = out_size)
  const float* W1     = (const float*)d_in[4];   // [1,128]
  const float* a_src1 = (const float*)d_in[5];   // [4,32]
  const float* a_dst1 = (const float*)d_in[6];   // [4,32]
  const float* b1     = (const float*)d_in[7];   // [128]
  const float* W2     = (const float*)d_in[8];   // [128,32]
  const float* a_src2 = (const float*)d_in[9];   // [1,32]
  const float* a_dst2 = (const float*)d_in[10];  // [1,32]
  const float* b2     = (const float*)d_in[11];  // [32]
  const float* lin_w  = (const float*)d_in[12];  // [32,1]
  const float* lin_b  = (const float*)d_in[13];  // [1]
  float* out = (float*)d_out;

  const long long N = in_sizes[0];       // x is [N,1]
  const long long E = in_sizes[1] / 2;
  const int G = out_size;
  const long long ET = E + N;            // edges + self loops

  // -------- workspace layout (floats) --------
  float* w = (float*)d_ws;
  float* ws1  = w;                       // 4
  float* wd1  = w + 4;                   // 4
  float* m1   = w + 8;                   // 4N  -inf-init region start
  float* m2   = m1 + 4 * N;              // N   (contiguous with m1: 5N total)
  float* s1   = m2 + N;                  // 4N  zero-init region start
  float* t1   = s1 + 4 * N;              // 4N
  float* s2   = t1 + 4 * N;              // N
  float* out2 = s2 + N;                  // 32N
  float* gsum = out2 + 32 * N;           // 32G
  float* gcnt = gsum + 32LL * G;         // G
  const long long zero_len = 41 * N + 33LL * G;
  float* x1   = w + align4((gcnt + G) - w);    // 128N (16B aligned)
  float* h2   = x1 + 128 * N;            // 32N
  float* es2  = h2 + 32 * N;             // N
  float* ed2  = es2 + N;                 // N
  (void)ws_size; (void)n_in;

  const int B = 256;

  // init: -inf for maxes (kernel), zeros for accumulators (single memset)
  init_neginf_kernel<<<cdiv_i(5 * N, B), B, 0, stream>>>(m1, 5 * N);
  hipMemsetAsync((void*)s1, 0, (size_t)zero_len * sizeof(float), stream);
  prep1_kernel<<<1, 32, 0, stream>>>(W1, a_src1, a_dst1, ws1, wd1);

  // ---- layer 1 (factorized: F_in == 1) ----
  edge1_max_kernel<<<cdiv_i(ET, B), B, 0, stream>>>(ei, x, ws1, wd1, m1, (int)E, (int)ET);
  edge1_sum_kernel<<<cdiv_i(ET, B), B, 0, stream>>>(ei, x, ws1, wd1, m1, s1, (int)E, (int)ET);
  edge1_agg_kernel<<<cdiv_i(ET, B), B, 0, stream>>>(ei, x, ws1, wd1, m1, s1, t1, (int)E, (int)ET);
  build_x1_kernel<<<cdiv_i(N * 32, B), B, 0, stream>>>(t1, W1, b1, x1, (int)N);

  // ---- layer 2 dense: h2 = x1 @ W2 via V_WMMA_F32_16X16X4_F32 ----
  {
    long long nwaves = (N + 15) / 16;
    gemm_h2_kernel<<<cdiv_i(nwaves * 32, B), B, 0, stream>>>(x1, W2, h2, (int)N);
  }
  attn2_kernel<<<cdiv_i(N, B), B, 0, stream>>>(h2, a_src2, a_dst2, es2, ed2, (int)N);

  // ---- layer 2 edge softmax + aggregation ----
  edge2_max_kernel<<<cdiv_i(ET, B), B, 0, stream>>>(ei, es2, ed2, m2, (int)E, (int)ET);
  edge2_sum_kernel<<<cdiv_i(ET, B), B, 0, stream>>>(ei, es2, ed2, m2, s2, (int)E, (int)ET);
  edge2_agg_kernel<<<cdiv_i(ET * 8, B), B, 0, stream>>>(ei, es2, ed2, m2, s2, h2, out2,
                                                        (int)E, (int)ET);

  // ---- pooling + readout ----
  pool_kernel<<<cdiv_i(N * 32, B), B, 0, stream>>>(out2, b2, batch, gsum, gcnt, (int)N);
  final_kernel<<<cdiv_i(G, 64), 64, 0, stream>>>(gsum, gcnt, lin_w, lin_b, out, G);
}